// AttentionLayer_79963701117362
// MI455X (gfx1250) — compile-verified
//
#include <hip/hip_runtime.h>
#include <math.h>

typedef __attribute__((ext_vector_type(2))) float v2f;
typedef __attribute__((ext_vector_type(8))) float v8f;

constexpr int SEQ  = 8192;  // N
constexpr int BDIM = 64;    // B (projected-away dim)

// ---------------------------------------------------------------------------
// Kernel A: projections qp/kp/vp[n] = sum_b X[b,n]*W[b] + bias
// via V_WMMA_F32_16X16X4_F32. One wave per 16 sequence positions.
// A tile  (16x4, MxK): lane<16 -> {K=0,1}, lane>=16 -> {K=2,3}, M = lane&15
// B tile  (4x16, KxN): vgpr0: lanes0-15 row K=0, lanes16-31 row K=2
//                      vgpr1: lanes0-15 row K=1, lanes16-31 row K=3
//   -> column N=0 lives entirely in lane 0 ({K0,K1}) and lane 16 ({K2,K3})
// C tile  (16x16):     vgpr r: lanes0-15 M=r, lanes16-31 M=r+8; column 0 is
//                      lane 0 (M=0..7) and lane 16 (M=8..15)
// ---------------------------------------------------------------------------
__global__ __launch_bounds__(32) void proj_wmma_kernel(
    const float* __restrict__ q, const float* __restrict__ k,
    const float* __restrict__ v,
    const float* __restrict__ Wq, const float* __restrict__ bq,
    const float* __restrict__ Wk, const float* __restrict__ bk,
    const float* __restrict__ Wv, const float* __restrict__ bv,
    float* __restrict__ qp, float* __restrict__ kp, float* __restrict__ vp)
{
    const int lane = threadIdx.x;        // 0..31 (wave32)
    const int n0   = blockIdx.x * 16;    // 16 sequence positions per wave
    const int m    = lane & 15;
    const int kk   = (lane >> 4) * 2;    // 0 for lanes 0-15, 2 for lanes 16-31

    v8f cq = {}; v8f ck = {}; v8f cv = {};

    #pragma unroll
    for (int kb = 0; kb < BDIM; kb += 4) {
        const int r0 = (kb + kk) * SEQ + n0 + m;
        v2f aq, ak, av;
        aq.x = q[r0]; aq.y = q[r0 + SEQ];
        ak.x = k[r0]; ak.y = k[r0 + SEQ];
        av.x = v[r0]; av.y = v[r0 + SEQ];

        v2f bqm = {0.f, 0.f}, bkm = {0.f, 0.f}, bvm = {0.f, 0.f};
        if (m == 0) {  // lanes 0 and 16 carry column 0 of B
            bqm.x = Wq[kb + kk]; bqm.y = Wq[kb + kk + 1];
            bkm.x = Wk[kb + kk]; bkm.y = Wk[kb + kk + 1];
            bvm.x = Wv[kb + kk]; bvm.y = Wv[kb + kk + 1];
        }
        cq = __builtin_amdgcn_wmma_f32_16x16x4_f32(false, aq, false, bqm,
                                                   (short)0, cq, false, false);
        ck = __builtin_amdgcn_wmma_f32_16x16x4_f32(false, ak, false, bkm,
                                                   (short)0, ck, false, false);
        cv = __builtin_amdgcn_wmma_f32_16x16x4_f32(false, av, false, bvm,
                                                   (short)0, cv, false, false);
    }

    if (m == 0) {  // lane 0: M=0..7, lane 16: M=8..15 of column 0
        const int base = n0 + (lane >> 4) * 8;
        const float bqs = bq[0], bks = bk[0], bvs = bv[0];
        #pragma unroll
        for (int r = 0; r < 8; ++r) {
            qp[base + r] = cq[r] + bqs;
            kp[base + r] = ck[r] + bks;
            vp[base + r] = cv[r] + bvs;
        }
    }
}

// ---------------------------------------------------------------------------
// Kernel B: rank-1 softmax rows + out = attn @ vp.
// kp/vp cached in LDS (64 KB). 8 waves/block, 2 rows per wave, 512 blocks.
// Non-temporal stores for the 256 MB attn stream (don't thrash L2).
// ---------------------------------------------------------------------------
__device__ __forceinline__ float wave_sum(float x) {
    #pragma unroll
    for (int off = 16; off > 0; off >>= 1) x += __shfl_xor(x, off, 32);
    return x;
}
__device__ __forceinline__ float wave_max(float x) {
    #pragma unroll
    for (int off = 16; off > 0; off >>= 1) x = fmaxf(x, __shfl_xor(x, off, 32));
    return x;
}
__device__ __forceinline__ float wave_min(float x) {
    #pragma unroll
    for (int off = 16; off > 0; off >>= 1) x = fminf(x, __shfl_xor(x, off, 32));
    return x;
}

__global__ __launch_bounds__(256) void softmax_rank1_kernel(
    const float* __restrict__ qp, const float* __restrict__ kp,
    const float* __restrict__ vp,
    float* __restrict__ out, float* __restrict__ attn)
{
    __shared__ float kp_s[SEQ];
    __shared__ float vp_s[SEQ];

    const int tid  = threadIdx.x;
    const int lane = tid & 31;
    const int wave = tid >> 5;          // 0..7

    // Cooperative LDS fill (float4 = 128-bit loads/stores)
    {
        const float4* kg = (const float4*)kp;
        const float4* vg = (const float4*)vp;
        float4* ks = (float4*)kp_s;
        float4* vs = (float4*)vp_s;
        #pragma unroll
        for (int j = tid; j < SEQ / 4; j += 256) { ks[j] = kg[j]; vs[j] = vg[j]; }
    }
    __syncthreads();

    // Per-wave global min/max of kp (exact row max of the rank-1 scores)
    float kmx = -INFINITY, kmn = INFINITY;
    for (int j = lane; j < SEQ; j += 32) {
        const float kv = kp_s[j];
        kmx = fmaxf(kmx, kv);
        kmn = fminf(kmn, kv);
    }
    kmx = wave_max(kmx);
    kmn = wave_min(kmn);

    const int row0 = blockIdx.x * 16 + wave * 2;

    #pragma unroll
    for (int rr = 0; rr < 2; ++rr) {
        const int i = row0 + rr;
        const float a    = qp[i] * 0.125f;               // 1/sqrt(64)
        const float mrow = (a >= 0.f) ? a * kmx : a * kmn;

        // Pass 1: denom and numerator
        float d = 0.f, num = 0.f;
        #pragma unroll 4
        for (int j = lane; j < SEQ; j += 32) {
            const float e = __expf(fmaf(a, kp_s[j], -mrow));
            d   += e;
            num += e * vp_s[j];
        }
        d   = wave_sum(d);
        num = wave_sum(num);
        const float inv = 1.0f / d;

        // Pass 2: write normalized row (streaming, non-temporal)
        float* __restrict__ arow = attn + (size_t)i * SEQ;
        #pragma unroll 4
        for (int j = lane; j < SEQ; j += 32) {
            const float e = __expf(fmaf(a, kp_s[j], -mrow));
            __builtin_nontemporal_store(e * inv, &arow[j]);
        }
        if (lane == 0) out[i] = num * inv;
    }
}

// ---------------------------------------------------------------------------
extern "C" void kernel_launch(void* const* d_in, const int* in_sizes, int n_in,
                              void* d_out, int out_size, void* d_ws, size_t ws_size,
                              hipStream_t stream) {
    (void)in_sizes; (void)n_in; (void)out_size; (void)ws_size;
    const float* q  = (const float*)d_in[0];
    const float* k  = (const float*)d_in[1];
    const float* v  = (const float*)d_in[2];
    const float* Wq = (const float*)d_in[3];
    const float* bq = (const float*)d_in[4];
    const float* Wk = (const float*)d_in[5];
    const float* bk = (const float*)d_in[6];
    const float* Wv = (const float*)d_in[7];
    const float* bv = (const float*)d_in[8];

    float* out  = (float*)d_out;        // [SEQ]
    float* attn = out + SEQ;            // [SEQ, SEQ], concatenated after `out`

    float* qp = (float*)d_ws;           // 3 * SEQ floats of scratch
    float* kp = qp + SEQ;
    float* vp = kp + SEQ;

    proj_wmma_kernel<<<SEQ / 16, 32, 0, stream>>>(q, k, v, Wq, bq, Wk, bk,
                                                  Wv, bv, qp, kp, vp);
    softmax_rank1_kernel<<<SEQ / 16, 256, 0, stream>>>(qp, kp, vp, out, attn);
}